// Pool_15135464751225
// MI455X (gfx1250) — compile-verified
//
#include <hip/hip_runtime.h>

typedef unsigned int uint32;
typedef __attribute__((ext_vector_type(2))) float v2f;
typedef __attribute__((ext_vector_type(8))) float v8f;

#define B   4
#define N   32768
#define D   512
#define K   8192

#define SORT_T 512          // threads per sort block (16 waves)
#define ELEMS  (N / SORT_T) // 64 contiguous elements per thread

// ---------------------------------------------------------------------------
// Kernel 1: logits[b][n] = dot(h[b][n][:], sf[b][:])  via V_WMMA_F32_16X16X4_F32
// Grid: 256 blocks/batch * 4 batches; block = 256 thr = 8 waves; wave = 16 rows.
// B-matrix columns all replicated with sf -> every column of D equals the score.
// ---------------------------------------------------------------------------
__global__ __launch_bounds__(256) void score_kernel(const float* __restrict__ h,
                                                    const float* __restrict__ sf,
                                                    float* __restrict__ logits) {
    __shared__ float sfs[D];
    const int b        = blockIdx.x >> 8;     // 256 blocks per batch
    const int blk_in_b = blockIdx.x & 255;

    for (int i = threadIdx.x; i < D; i += 256) sfs[i] = sf[b * D + i];
    __syncthreads();

    const int wave     = threadIdx.x >> 5;
    const int lane     = threadIdx.x & 31;
    const int row_base = blk_in_b * 128 + wave * 16;      // row within batch
    const int m        = lane & 15;
    const int khalf    = (lane >> 4) * 2;                 // lanes 0-15: K={0,1}; 16-31: K={2,3}

    const float* hrow = h + ((size_t)b * N + row_base + m) * D;

    v8f c = {};
    for (int kb = 0; kb < D; kb += 4) {
        v2f a  = *(const v2f*)(hrow + kb + khalf);        // A[m][kb+khalf .. +1]
        v2f bb = *(const v2f*)(sfs  + kb + khalf);        // B replicated across 16 cols
        c = __builtin_amdgcn_wmma_f32_16x16x4_f32(
                /*neg_a=*/false, a, /*neg_b=*/false, bb,
                /*c_mod=*/(short)0, c, /*reuse_a=*/false, /*reuse_b=*/false);
    }

    // C/D layout: VGPR r -> M=r (lanes 0-15), M=r+8 (lanes 16-31); take N=0 column.
    if ((lane & 15) == 0) {
        float* lg = logits + (size_t)b * N + row_base + (lane >> 4) * 8;
        lg[0] = c[0]; lg[1] = c[1]; lg[2] = c[2]; lg[3] = c[3];
        lg[4] = c[4]; lg[5] = c[5]; lg[6] = c[6]; lg[7] = c[7];
    }
}

// ---------------------------------------------------------------------------
// Kernel 2: build sortable keys. Ascending key == descending logit, so a
// stable ascending sort reproduces jax.lax.top_k order (ties: lower idx first).
// ---------------------------------------------------------------------------
__global__ __launch_bounds__(256) void keyinit_kernel(const float* __restrict__ logits,
                                                      uint32* __restrict__ keyA,
                                                      uint32* __restrict__ idxA) {
    const int i = blockIdx.x * 256 + threadIdx.x;   // 0 .. B*N-1
    uint32 u    = __float_as_uint(logits[i]);
    uint32 kasc = (u & 0x80000000u) ? ~u : (u | 0x80000000u);
    keyA[i] = ~kasc;
    idxA[i] = (uint32)(i & (N - 1));                // index within batch
}

// ---------------------------------------------------------------------------
// Kernel 3: ONE stable LSD radix pass (4-bit digit at `shift`), src -> dst.
// One 512-thread workgroup per batch; each thread owns 64 contiguous elements.
// Per-thread bin counters live in the thread's own column of LDS flat[16][512]
// (34KB static LDS). Pure integer kernel; launched 8 times with swapped bufs.
// ---------------------------------------------------------------------------
__global__ __launch_bounds__(SORT_T) void radix_pass_kernel(const uint32* __restrict__ skA,
                                                            const uint32* __restrict__ siA,
                                                            uint32* __restrict__ dkA,
                                                            uint32* __restrict__ diA,
                                                            int shift) {
    __shared__ uint32 flat[16 * SORT_T];   // bin-major counts -> offsets -> run ctrs
    __shared__ uint32 part[SORT_T];

    const int b = blockIdx.x;
    const int t = threadIdx.x;
    const uint32* sk = skA + (size_t)b * N;
    const uint32* si = siA + (size_t)b * N;
    uint32*       dk = dkA + (size_t)b * N;
    uint32*       di = diA + (size_t)b * N;

    const int base = t * ELEMS;
    const int fb   = t * 16;

    // 1) per-thread digit histogram into own LDS column (no contention)
    #pragma unroll
    for (int d = 0; d < 16; ++d) flat[d * SORT_T + t] = 0u;
    for (int i = 0; i < ELEMS; ++i) {
        uint32 dig = (sk[base + i] >> shift) & 15u;
        flat[dig * SORT_T + t] += 1u;
    }
    __syncthreads();

    // 2) exclusive scan of flat[] (bin-major) -> stable base offsets
    uint32 seg = 0;
    #pragma unroll
    for (int j = 0; j < 16; ++j) {
        uint32 v = flat[fb + j];
        flat[fb + j] = seg;            // exclusive within segment
        seg += v;
    }
    part[t] = seg;
    __syncthreads();
    for (int off = 1; off < SORT_T; off <<= 1) {   // inclusive Hillis-Steele
        uint32 v = (t >= off) ? part[t - off] : 0u;
        __syncthreads();
        part[t] += v;
        __syncthreads();
    }
    const uint32 segoff = part[t] - seg;            // exclusive block offset
    #pragma unroll
    for (int j = 0; j < 16; ++j) flat[fb + j] += segoff;
    __syncthreads();

    // 3) stable scatter; running counters are the thread's own LDS column
    for (int i = 0; i < ELEMS; ++i) {
        uint32 k   = sk[base + i];
        uint32 id  = si[base + i];
        uint32 d   = (k >> shift) & 15u;
        uint32 dst = flat[d * SORT_T + t];
        flat[d * SORT_T + t] = dst + 1u;
        dk[dst] = k;
        di[dst] = id;
    }
}

// ---------------------------------------------------------------------------
// Kernel 4: out[b][j][:] = h[b][sidx[j]][:] * sigmoid(logit[sidx[j]])
// sidx = first K entries of the fully sorted (descending-score) index array.
// One wave per output row; float4 (b128) streaming.
// ---------------------------------------------------------------------------
__global__ __launch_bounds__(256) void gather_kernel(const float* __restrict__ h,
                                                     const uint32* __restrict__ idxA,
                                                     const float* __restrict__ logits,
                                                     float* __restrict__ out) {
    const int wave = threadIdx.x >> 5;
    const int lane = threadIdx.x & 31;
    const size_t r = (size_t)blockIdx.x * 8 + wave;   // 0 .. B*K-1
    const int b = (int)(r >> 13);                      // K = 8192 = 2^13
    const int j = (int)(r & (K - 1));

    const uint32 id = idxA[(size_t)b * N + j];
    const float  x  = logits[(size_t)b * N + id];
    const float  v  = 1.0f / (1.0f + __expf(-x));

    const float4* src = (const float4*)(h + ((size_t)b * N + id) * D);
    float4*       dst = (float4*)(out + r * D);
    for (int i = lane; i < D / 4; i += 32) {
        float4 x4 = src[i];
        x4.x *= v; x4.y *= v; x4.z *= v; x4.w *= v;
        dst[i] = x4;
    }
}

// ---------------------------------------------------------------------------
extern "C" void kernel_launch(void* const* d_in, const int* in_sizes, int n_in,
                              void* d_out, int out_size, void* d_ws, size_t ws_size,
                              hipStream_t stream) {
    const float* h  = (const float*)d_in[0];   // [4][32768][512] f32
    const float* sf = (const float*)d_in[1];   // [4][1][512]     f32
    float* out = (float*)d_out;                // [4][8192][512]  f32

    // Workspace layout (2.5 MB total)
    char* ws = (char*)d_ws;
    float*  logits = (float*)(ws);                          // 512 KB
    uint32* keyA   = (uint32*)(ws + (512u << 10));          // 512 KB
    uint32* idxA   = (uint32*)(ws + (1024u << 10));         // 512 KB
    uint32* keyB   = (uint32*)(ws + (1536u << 10));         // 512 KB
    uint32* idxB   = (uint32*)(ws + (2048u << 10));         // 512 KB

    score_kernel<<<B * 256, 256, 0, stream>>>(h, sf, logits);
    keyinit_kernel<<<(B * N) / 256, 256, 0, stream>>>(logits, keyA, idxA);

    // 8 stable 4-bit passes, ping-ponging A <-> B; ends back in A.
    for (int pass = 0; pass < 8; ++pass) {
        if ((pass & 1) == 0)
            radix_pass_kernel<<<B, SORT_T, 0, stream>>>(keyA, idxA, keyB, idxB, pass * 4);
        else
            radix_pass_kernel<<<B, SORT_T, 0, stream>>>(keyB, idxB, keyA, idxA, pass * 4);
    }

    gather_kernel<<<(B * K) / 8, 256, 0, stream>>>(h, idxA, logits, out);
}